// SleepBandEnhancerPlusPlus_50723563765870
// MI455X (gfx1250) — compile-verified
//
#include <hip/hip_runtime.h>
#include <math.h>

typedef __attribute__((ext_vector_type(2))) float v2f;
typedef __attribute__((ext_vector_type(8))) float v8f;

#define NBANDS 6
#define C3 3
#define NF 6
#define T_DIM 3000
#define F_DIM 103
#define B_DIM 64
#define TILE_T 64
#define BLOCK_THREADS 128

// band tables (compile-time, from reference BANDS)
__device__ __constant__ int dBS[NBANDS]   = {1, 10, 20, 30, 40, 61};
__device__ __constant__ int dNB[NBANDS]   = {10, 11, 11, 11, 22, 16};
__device__ __constant__ int dFPAD[NBANDS] = {12, 12, 12, 12, 24, 16};
__device__ __constant__ int dBOFF[NBANDS] = {0, 576, 1152, 1728, 2304, 3456};
#define BW_TOT 4224   // sum over bands of 3*FPAD*16

__device__ __forceinline__ float softplus_f(float x) {
    return fmaxf(x, 0.0f) + log1pf(expf(-fabsf(x)));
}
__device__ __forceinline__ float gelu_f(float x) {
    return 0.5f * x * (1.0f + erff(x * 0.70710678118654752f));
}
__device__ __forceinline__ float sigmoid_f(float x) {
    return 1.0f / (1.0f + expf(-x));
}

// ---------------------------------------------------------------------------
// Kernel 1: precompute normalized Gaussian filter weights wnorm[i][c][n][f]
// stride 24 in f.  108 (i,c,n) triples.
// ---------------------------------------------------------------------------
__global__ void wnorm_setup_kernel(const float* __restrict__ centers,
                                   const float* __restrict__ widths,
                                   const float* __restrict__ gains,
                                   float* __restrict__ wnorm) {
    int id = threadIdx.x;
    if (id >= NBANDS * C3 * NF) return;
    int i = id / (C3 * NF);
    int s = dBS[i];
    int nb = dNB[i];
    float mu = softplus_f(centers[id]) + (float)s;
    mu = fminf(fmaxf(mu, (float)s), (float)(s + nb - 1));
    float sd = softplus_f(widths[id]) + 0.001f;
    sd = fminf(fmaxf(sd, 0.5f), 2.0f * (float)nb / 6.0f);
    float g[24];
    float sum = 0.0f;
    for (int f = 0; f < nb; ++f) {
        float z = ((float)f - mu) / sd;
        float e = expf(-0.5f * z * z);
        g[f] = e;
        sum += e;
    }
    float scale = gains[id] / (sum + 1e-6f);
    for (int f = 0; f < 24; ++f)
        wnorm[id * 24 + f] = (f < nb) ? g[f] * scale : 0.0f;
}

// ---------------------------------------------------------------------------
// Kernel 2: fused forward.  Block = 128 threads (4 waves), covers one
// (b, 64-row t tile).  Phase A uses V_WMMA_F32_16X16X4_F32 for band filtering.
// ---------------------------------------------------------------------------
__global__ void __launch_bounds__(BLOCK_THREADS)
band_enhancer_kernel(const float* __restrict__ spec,
                     const float* __restrict__ wnorm,
                     const float* __restrict__ align_w,
                     const float* __restrict__ align_b,
                     const float* __restrict__ fc1_w,
                     const float* __restrict__ fc1_b,
                     const float* __restrict__ fc2_w,
                     const float* __restrict__ fc2_b,
                     const float* __restrict__ band_gain,
                     const float* __restrict__ gate_w,
                     const float* __restrict__ gate_b,
                     const float* __restrict__ proj_w,
                     const float* __restrict__ proj_b,
                     const float* __restrict__ bn_gamma,
                     const float* __restrict__ bn_beta,
                     const float* __restrict__ bn_mean,
                     const float* __restrict__ bn_var,
                     float* __restrict__ out,
                     int num_tiles) {
    __shared__ float sBw[BW_TOT];           // zero-padded B matrices [band][c][fpad][16]
    __shared__ float sBS[TILE_T * 108];     // band_stack tile [t][i][d][n]
    __shared__ float sAlignW[54];
    __shared__ float sAlignB[18];
    __shared__ float sFc1W[108];
    __shared__ float sFc1B[6];
    __shared__ float sFc2W[108];
    __shared__ float sFc2B[18];
    __shared__ float sBGain[6];
    __shared__ float sGateW[9];
    __shared__ float sGateB[3];
    __shared__ float sProjW[9];
    __shared__ float sProjB[3];
    __shared__ float sBnSc[3];
    __shared__ float sBnSh[3];

    const int tid = threadIdx.x;
    const int b = blockIdx.x / num_tiles;
    const int tile = blockIdx.x % num_tiles;
    const int tbase = tile * TILE_T;

    // ---- load small params ----
    for (int i = tid; i < 54; i += BLOCK_THREADS) sAlignW[i] = align_w[i];
    for (int i = tid; i < 18; i += BLOCK_THREADS) sAlignB[i] = align_b[i];
    for (int i = tid; i < 108; i += BLOCK_THREADS) sFc1W[i] = fc1_w[i];
    for (int i = tid; i < 6; i += BLOCK_THREADS) sFc1B[i] = fc1_b[i];
    for (int i = tid; i < 108; i += BLOCK_THREADS) sFc2W[i] = fc2_w[i];
    for (int i = tid; i < 18; i += BLOCK_THREADS) sFc2B[i] = fc2_b[i];
    for (int i = tid; i < 6; i += BLOCK_THREADS) sBGain[i] = band_gain[i];
    for (int i = tid; i < 9; i += BLOCK_THREADS) sGateW[i] = gate_w[i];
    for (int i = tid; i < 3; i += BLOCK_THREADS) sGateB[i] = gate_b[i];
    for (int i = tid; i < 9; i += BLOCK_THREADS) sProjW[i] = proj_w[i];
    for (int i = tid; i < 3; i += BLOCK_THREADS) sProjB[i] = proj_b[i];
    if (tid < 3) {
        float sc = bn_gamma[tid] * rsqrtf(bn_var[tid] + 1e-5f);
        sBnSc[tid] = sc;
        sBnSh[tid] = bn_beta[tid] - bn_mean[tid] * sc;
    }
    // ---- build zero-padded B matrices in LDS: sBw[boff + (c*fpad + f)*16 + n]
    for (int idx = tid; idx < BW_TOT; idx += BLOCK_THREADS) {
        int i = 0, rem = idx;
        while (i < NBANDS - 1 && rem >= 3 * dFPAD[i] * 16) { rem -= 3 * dFPAD[i] * 16; ++i; }
        int fp = dFPAD[i];
        int c = rem / (fp * 16);
        int r2 = rem - c * fp * 16;
        int f = r2 >> 4;
        int n = r2 & 15;
        float v = 0.0f;
        if (n < NF && f < dNB[i]) v = wnorm[((i * 3 + c) * NF + n) * 24 + f];
        sBw[idx] = v;
    }
    __syncthreads();

    // =========================== Phase A: WMMA filtering =====================
    {
        const int lane = tid & 31;
        const int wv = tid >> 5;
        const int m = lane & 15;          // t-row within 16 (A) / column n (B, D)
        const int hi2 = (lane >> 4) << 1; // K sub-offset: 0 for lanes 0-15, 2 for 16-31
        int tg = tbase + wv * 16 + m;
        if (tg > T_DIM - 1) tg = T_DIM - 1;
        const float* rows[3];
        rows[0] = spec + ((size_t)((b * 3 + 0) * T_DIM) + tg) * F_DIM;
        rows[1] = spec + ((size_t)((b * 3 + 1) * T_DIM) + tg) * F_DIM;
        rows[2] = spec + ((size_t)((b * 3 + 2) * T_DIM) + tg) * F_DIM;

#pragma unroll
        for (int i = 0; i < NBANDS; ++i) {
            v8f filt[3];
#pragma unroll
            for (int c = 0; c < 3; ++c) {
                v8f acc = {0.f, 0.f, 0.f, 0.f, 0.f, 0.f, 0.f, 0.f};
                const float* rp = rows[c] + dBS[i];
                const float* bb = &sBw[dBOFF[i] + c * dFPAD[i] * 16];
                const int nchunks = dFPAD[i] >> 2;
                for (int kk = 0; kk < nchunks; ++kk) {
                    int lf = kk * 4 + hi2;
                    v2f a;
                    a[0] = rp[lf];
                    a[1] = rp[lf + 1];
                    v2f bm;
                    bm[0] = bb[lf * 16 + m];
                    bm[1] = bb[(lf + 1) * 16 + m];
                    acc = __builtin_amdgcn_wmma_f32_16x16x4_f32(
                        false, a, false, bm, (short)0, acc, false, false);
                }
                filt[c] = acc;
            }
            // fold 3x3 align in the D layout (lane = column n, vgpr = t row),
            // scatter to sBS[t][i][d][n]
            if (m < NF) {
#pragma unroll
                for (int d = 0; d < 3; ++d) {
                    float aw0 = sAlignW[i * 9 + d * 3 + 0];
                    float aw1 = sAlignW[i * 9 + d * 3 + 1];
                    float aw2 = sAlignW[i * 9 + d * 3 + 2];
                    float ab = sAlignB[i * 3 + d];
#pragma unroll
                    for (int v = 0; v < 8; ++v) {
                        int tl = wv * 16 + ((lane < 16) ? v : v + 8);
                        float val = filt[0][v] * aw0 + filt[1][v] * aw1 +
                                    filt[2][v] * aw2 + ab;
                        sBS[tl * 108 + i * 18 + d * 6 + m] = val;
                    }
                }
            }
        }
    }
    __syncthreads();

    // ====================== Phase attention: one thread per t ================
    if (tid < TILE_T) {
        float* bsT = &sBS[tid * 108];   // [k=i*3+d][n] contiguous, k*6+n
        float h[6][6];
        for (int o = 0; o < 6; ++o) {
            for (int n = 0; n < 6; ++n) {
                float s = sFc1B[o];
#pragma unroll
                for (int k = 0; k < 18; ++k)
                    s += sFc1W[o * 18 + k] * bsT[k * 6 + n];
                h[o][n] = gelu_f(s);
            }
        }
        for (int d = 0; d < 3; ++d) {
            for (int n = 0; n < 6; ++n) {
                float l[6];
                float mx = -3.402823e38f;
#pragma unroll
                for (int i = 0; i < 6; ++i) {
                    int k = i * 3 + d;
                    float s = sFc2B[k];
#pragma unroll
                    for (int o = 0; o < 6; ++o) s += sFc2W[k * 6 + o] * h[o][n];
                    l[i] = s;
                    mx = fmaxf(mx, s);
                }
                float sum = 0.0f;
#pragma unroll
                for (int i = 0; i < 6; ++i) { l[i] = expf(l[i] - mx); sum += l[i]; }
                float inv = 1.0f / sum;
#pragma unroll
                for (int i = 0; i < 6; ++i)
                    bsT[(i * 3 + d) * 6 + n] *= l[i] * inv * sBGain[i];
            }
        }
    }
    __syncthreads();

    // =============== Phase B: interp + gate + proj + BN + GELU ===============
    {
        const int t_local = tid >> 1;
        const int half = tid & 1;
        const int t = tbase + t_local;
        if (t < T_DIM) {
            const int f0 = half ? 52 : 0;
            const int f1 = half ? F_DIM : 52;
            const float* sp0 = spec + ((size_t)(b * 3 + 0) * T_DIM + t) * F_DIM;
            const float* sp1 = spec + ((size_t)(b * 3 + 1) * T_DIM + t) * F_DIM;
            const float* sp2 = spec + ((size_t)(b * 3 + 2) * T_DIM + t) * F_DIM;
            float* op0 = out + ((size_t)(b * 3 + 0) * T_DIM + t) * F_DIM;
            float* op1 = out + ((size_t)(b * 3 + 1) * T_DIM + t) * F_DIM;
            float* op2 = out + ((size_t)(b * 3 + 2) * T_DIM + t) * F_DIM;
            const float* bsT = &sBS[t_local * 108];

            for (int f = f0; f < f1; ++f) {
                float e0 = 0.f, e1 = 0.f, e2v = 0.f;
#pragma unroll
                for (int i = 0; i < NBANDS; ++i) {
                    int s = dBS[i], nb = dNB[i];
                    if (f >= s && f <= s + nb - 1) {
                        int j = f - s;
                        float src = ((float)j + 0.5f) * 6.0f / (float)nb - 0.5f;
                        src = fmaxf(src, 0.0f);
                        int x0 = (int)src;
                        int x1 = (x0 + 1 < 5) ? x0 + 1 : 5;
                        float wl = src - (float)x0;
                        float w0 = 1.0f - wl;
                        const float* p = &bsT[i * 18];
                        e0 += p[x0] * w0 + p[x1] * wl;
                        e1 += p[6 + x0] * w0 + p[6 + x1] * wl;
                        e2v += p[12 + x0] * w0 + p[12 + x1] * wl;
                    }
                }
                float s0 = sp0[f], s1 = sp1[f], s2 = sp2[f];
                float r0 = e0 - s0, r1 = e1 - s1, r2 = e2v - s2;
                float g0 = sigmoid_f(sGateB[0] + sGateW[0] * r0 + sGateW[1] * r1 + sGateW[2] * r2);
                float g1 = sigmoid_f(sGateB[1] + sGateW[3] * r0 + sGateW[4] * r1 + sGateW[5] * r2);
                float g2 = sigmoid_f(sGateB[2] + sGateW[6] * r0 + sGateW[7] * r1 + sGateW[8] * r2);
                float q0 = s0 + g0 * r0;
                float q1 = s1 + g1 * r1;
                float q2 = s2 + g2 * r2;
                float y0 = sProjB[0] + sProjW[0] * q0 + sProjW[1] * q1 + sProjW[2] * q2;
                float y1 = sProjB[1] + sProjW[3] * q0 + sProjW[4] * q1 + sProjW[5] * q2;
                float y2 = sProjB[2] + sProjW[6] * q0 + sProjW[7] * q1 + sProjW[8] * q2;
                y0 = y0 * sBnSc[0] + sBnSh[0];
                y1 = y1 * sBnSc[1] + sBnSh[1];
                y2 = y2 * sBnSc[2] + sBnSh[2];
                op0[f] = gelu_f(y0);
                op1[f] = gelu_f(y1);
                op2[f] = gelu_f(y2);
            }
        }
    }
}

// ---------------------------------------------------------------------------
extern "C" void kernel_launch(void* const* d_in, const int* in_sizes, int n_in,
                              void* d_out, int out_size, void* d_ws, size_t ws_size,
                              hipStream_t stream) {
    const float* spec      = (const float*)d_in[0];
    const float* centers   = (const float*)d_in[1];
    const float* widths    = (const float*)d_in[2];
    const float* gains     = (const float*)d_in[3];
    const float* align_w   = (const float*)d_in[4];
    const float* align_b   = (const float*)d_in[5];
    const float* fc1_w     = (const float*)d_in[6];
    const float* fc1_b     = (const float*)d_in[7];
    const float* fc2_w     = (const float*)d_in[8];
    const float* fc2_b     = (const float*)d_in[9];
    const float* band_gain = (const float*)d_in[10];
    const float* gate_w    = (const float*)d_in[11];
    const float* gate_b    = (const float*)d_in[12];
    const float* proj_w    = (const float*)d_in[13];
    const float* proj_b    = (const float*)d_in[14];
    const float* bn_gamma  = (const float*)d_in[15];
    const float* bn_beta   = (const float*)d_in[16];
    const float* bn_mean   = (const float*)d_in[17];
    const float* bn_var    = (const float*)d_in[18];
    float* out = (float*)d_out;
    float* wnorm = (float*)d_ws;   // 6*3*6*24 floats = 10368 bytes

    wnorm_setup_kernel<<<1, 128, 0, stream>>>(centers, widths, gains, wnorm);

    const int num_tiles = (T_DIM + TILE_T - 1) / TILE_T;   // 47
    dim3 grid(B_DIM * num_tiles);
    band_enhancer_kernel<<<grid, BLOCK_THREADS, 0, stream>>>(
        spec, wnorm, align_w, align_b, fc1_w, fc1_b, fc2_w, fc2_b,
        band_gain, gate_w, gate_b, proj_w, proj_b,
        bn_gamma, bn_beta, bn_mean, bn_var, out, num_tiles);
}